// Atomistic_51702816309466
// MI455X (gfx1250) — compile-verified
//
#include <hip/hip_runtime.h>

typedef __attribute__((ext_vector_type(2))) float v2f;
typedef __attribute__((ext_vector_type(4))) float v4f;
typedef __attribute__((ext_vector_type(8))) float v8f;
typedef __attribute__((ext_vector_type(4))) int   v4i;

#define WAVES_PER_BLOCK 8
#define BLOCK_THREADS 256
#define MAX_BLOCKS 2048
#define STAGE_F 2048   // 32 atoms * 64 feats per wave
#define DTILE_F 512    // two 16x16 f32 D tiles per wave

#if defined(__gfx1250__) && __has_builtin(__builtin_amdgcn_global_load_async_to_lds_b128)
#define HAVE_ASYNC_LDS 1
typedef __attribute__((address_space(1))) v4i* gptr_b128;
typedef __attribute__((address_space(3))) v4i* lptr_b128;
#else
#define HAVE_ASYNC_LDS 0
#endif

__global__ __launch_bounds__(BLOCK_THREADS) void atoms_wmma_kernel(
    const float* __restrict__ X,        // [nAtoms, 64]
    const float* __restrict__ W,        // [8, 64] (OUT_DIM = 1)
    const int* __restrict__ species,    // [nAtoms]
    const int* __restrict__ segids,     // [nAtoms] sorted
    float* __restrict__ out,            // [n_structures]
    int nAtoms)
{
    __shared__ float stage[WAVES_PER_BLOCK * STAGE_F];
    __shared__ float dtile[WAVES_PER_BLOCK * DTILE_F];

    const int lane = threadIdx.x & 31;
    const int waveInBlock = threadIdx.x >> 5;
    const int hi = lane >> 4;   // lane-half: adds 2 to K index in A/B fragments
    const int n  = lane & 15;   // B column / atom-in-tile
    float* ws = &stage[waveInBlock * STAGE_F];
    float* wd = &dtile[waveInBlock * DTILE_F];

    // ---- Preload B fragments once: Bt[k][col] = W[col][k], cols 8..15 = 0 ----
    // v0 holds row K = 4*kk + 2*hi, v1 holds K = 4*kk + 2*hi + 1 (mirrors A).
    v2f bfrag[16];
    #pragma unroll
    for (int kk = 0; kk < 16; ++kk) {
        float b0 = 0.0f, b1 = 0.0f;
        if (n < 8) {
            const float* wp = W + n * 64 + 4 * kk + 2 * hi;
            b0 = wp[0];
            b1 = wp[1];
        }
        bfrag[kk].x = b0;
        bfrag[kk].y = b1;
    }

    const int nChunks = (nAtoms + 31) >> 5;
    const int waveId  = blockIdx.x * WAVES_PER_BLOCK + waveInBlock;
    const int nWaves  = gridDim.x * WAVES_PER_BLOCK;
    const v4f* X4 = (const v4f*)X;
    const long f4Max = (long)nAtoms * 16 - 1;

    for (int chunk = waveId; chunk < nChunks; chunk += nWaves) {
        const int base = chunk << 5;

        // Prefetch next chunk into GL2 -> global_prefetch_b8
        {
            const int nextBase = base + (nWaves << 5);
            if (nextBase + 32 <= nAtoms) {
                const char* p =
                    (const char*)(X + (size_t)nextBase * 64) + (size_t)lane * 256;
                __builtin_prefetch(p, 0, 3);
                __builtin_prefetch(p + 128, 0, 3);
            }
        }

        // ---- Stage 8 KB of X (32 atom rows) into LDS, fully coalesced ----
        const long f4Base = (long)base * 16;
        #pragma unroll
        for (int i = 0; i < 16; ++i) {
            long f4 = f4Base + i * 32 + lane;
            if (f4 > f4Max) f4 = f4Max;           // tail clamp (masked later)
            float* dst = ws + (i * 32 + lane) * 4;
#if HAVE_ASYNC_LDS
            __builtin_amdgcn_global_load_async_to_lds_b128(
                (gptr_b128)(X4 + f4), (lptr_b128)dst, 0, 0);
#else
            *(v4f*)dst = X4[f4];
#endif
        }
#if HAVE_ASYNC_LDS
  #if __has_builtin(__builtin_amdgcn_s_wait_asynccnt)
        __builtin_amdgcn_s_wait_asynccnt(0);
        asm volatile("" ::: "memory");
  #else
        asm volatile("s_wait_asynccnt 0x0" ::: "memory");
  #endif
#else
        asm volatile("" ::: "memory");
#endif

        // ---- Two 16-atom WMMA tiles: D = Xtile(16x64) * Wt(64x16) ----
        #pragma unroll
        for (int t = 0; t < 2; ++t) {
            const float* xr = ws + (t * 16 + n) * 64 + 2 * hi;

            v2f a[16];                       // all A fragments up front (MLP)
            #pragma unroll
            for (int kk = 0; kk < 16; ++kk)
                a[kk] = *(const v2f*)(xr + 4 * kk);

            v8f c = {};
            #pragma unroll
            for (int kk = 0; kk < 16; ++kk)
                c = __builtin_amdgcn_wmma_f32_16x16x4_f32(
                        false, a[kk], false, bfrag[kk], (short)0, c, false, false);

            // D[M][N] -> wd[t*256 + (M&7)*32 + (M>>3)*16 + N]
            #pragma unroll
            for (int v = 0; v < 8; ++v)
                wd[t * 256 + v * 32 + lane] = c[v];
        }
        asm volatile("s_wait_dscnt 0x0" ::: "memory");  // same-wave LDS RAW

        // ---- Extract per-atom value: lane L -> atom base+L ----
        const int atomIdx = base + lane;
        float val = 0.0f;
        int st = -1;
        if (atomIdx < nAtoms) {
            const int s = species[atomIdx];
            const int m = lane & 15;
            const int t = lane >> 4;
            val = wd[t * 256 + (m & 7) * 32 + (m >> 3) * 16 + s];
            st = segids[atomIdx];
        }

        // ---- Segmented suffix-sum over 32 sorted lanes, heads do atomics ----
        #pragma unroll
        for (int off = 1; off < 32; off <<= 1) {
            float vo = __shfl_down(val, off, 32);
            int   so = __shfl_down(st,  off, 32);
            if ((lane + off < 32) && (so == st)) val += vo;
        }
        const int stUp = __shfl_up(st, 1, 32);
        const bool head = (st >= 0) && ((lane == 0) || (stUp != st));
        if (head) atomicAdd(&out[st], val);
    }
}

extern "C" void kernel_launch(void* const* d_in, const int* in_sizes, int n_in,
                              void* d_out, int out_size, void* d_ws, size_t ws_size,
                              hipStream_t stream) {
    const float* X       = (const float*)d_in[0];   // [N, 64] f32
    const float* W       = (const float*)d_in[1];   // [8, 64, 1] f32
    const int*   species = (const int*)d_in[2];     // [N] i32
    const int*   segids  = (const int*)d_in[3];     // [N] i32, sorted
    float*       out     = (float*)d_out;           // [n_structures, 1] f32

    const int nAtoms  = in_sizes[0] / 64;
    const int nChunks = (nAtoms + 31) / 32;
    int blocks = (nChunks + WAVES_PER_BLOCK - 1) / WAVES_PER_BLOCK;
    if (blocks > MAX_BLOCKS) blocks = MAX_BLOCKS;
    if (blocks < 1) blocks = 1;

    // Atomics accumulate: out must start at zero every call (capture-safe).
    (void)hipMemsetAsync(d_out, 0, (size_t)out_size * sizeof(float), stream);

    atoms_wmma_kernel<<<blocks, BLOCK_THREADS, 0, stream>>>(
        X, W, species, segids, out, nAtoms);
}